// NerfRenderer_56873956933822
// MI455X (gfx1250) — compile-verified
//
#include <hip/hip_runtime.h>
#include <hip/hip_bf16.h>

// ---------------------------------------------------------------------------
// NeRF render + distortion loss for MI455X (gfx1250, wave32).
//   inputs : sigmas [1024,576] f32, rgbs [1024,576,3] f32
//   outputs: image [1024,3] | invdepth [1024] | l_dist  (4097 f32, concat)
// Distortion term computed as tiled W @ S via V_WMMA_F32_16X16X4_F32 with the
// |z_i - z_j| tiles synthesized on the fly from LDS (S never materialized).
// ---------------------------------------------------------------------------

typedef __attribute__((ext_vector_type(2))) float v2f;
typedef __attribute__((ext_vector_type(8))) float v8f;

#define N_SAMP   576
#define N_RAYS   1024
#define N_TILES  36          // 576 / 16
#define N_GROUPS 64          // 1024 / 16
#define EPS      1e-10f

// workspace layout in floats
#define WS_ZVALS 0
#define WS_DELTA 576
#define WS_INVZ  1152
#define WS_ZLIN  1728
#define WS_W     2304                       // weights matrix [1024][576]
#define WS_PART  (WS_W + N_RAYS * N_SAMP)   // per-(ray,tile) partials [1024][36]
// total = 628992 floats = 2.46 MB of d_ws

// ---------------------------------------------------------------------------
// Kernel 1: build z tables. z_vals = 10^linspace, deltas, 1/z, normalized lin.
// ---------------------------------------------------------------------------
__global__ void k_init(float* ws) {
    float* zvals = ws + WS_ZVALS;
    float* delta = ws + WS_DELTA;
    float* invz  = ws + WS_INVZ;
    float* zlin  = ws + WS_ZLIN;

    for (int i = threadIdx.x; i < N_SAMP; i += blockDim.x) {
        float x;
        if (i < 384) x = -1.2f + 1.2f * (float)i / 383.0f;        // logspace exps
        else         x =  2.0f * (float)(i - 384) / 191.0f;
        float z = exp10f(x);
        zvals[i] = z;
        invz[i]  = 1.0f / z;
        zlin[i]  = (x + 1.2f) * (1.0f / 3.2f);                    // [0,1]
    }
    __syncthreads();
    for (int i = threadIdx.x; i < N_SAMP; i += blockDim.x)
        delta[i] = (i < N_SAMP - 1) ? (zvals[i + 1] - zvals[i]) : 1e10f;
}

// ---------------------------------------------------------------------------
// Kernel 2: per-ray transmittance scan -> weights, image, invdepth.
// One wave32 per ray; lane owns 18 contiguous samples. Exclusive cumprod of
// (exp(-sigma*delta) + EPS) via local products + __shfl_up wave scan.
// blockDim = 256 (8 waves/block), gridDim = 128.
// ---------------------------------------------------------------------------
__global__ void k_weights(const float* __restrict__ sigmas,
                          const float* __restrict__ rgbs,
                          const float* __restrict__ ws,
                          float* __restrict__ Wmat,
                          float* __restrict__ out) {
    const float* delta = ws + WS_DELTA;
    const float* invz  = ws + WS_INVZ;

    const int lane = threadIdx.x & 31;
    const int ray  = blockIdx.x * 8 + (threadIdx.x >> 5);
    const float* sig = sigmas + ray * N_SAMP;
    const float* rgb = rgbs   + ray * N_SAMP * 3;

    const int base = lane * 18;                     // 576 / 32 samples per lane
    float t[18];                                    // cumprod factors
    float lp = 1.0f;                                // product of this lane chunk
#pragma unroll
    for (int j = 0; j < 18; ++j) {
        int i = base + j;
        float e = expf(-sig[i] * delta[i]);         // = 1 - alpha
        t[j] = e + EPS;
        lp *= t[j];
    }
    // inclusive wave scan of chunk products, then shift to exclusive
    float sc = lp;
#pragma unroll
    for (int off = 1; off < 32; off <<= 1) {
        float n = __shfl_up(sc, off, 32);
        if (lane >= off) sc *= n;
    }
    float run = __shfl_up(sc, 1, 32);
    if (lane == 0) run = 1.0f;                      // exclusive transmittance

    float ir = 0.0f, ig = 0.0f, ib = 0.0f, idp = 0.0f;
#pragma unroll
    for (int j = 0; j < 18; ++j) {
        int i = base + j;
        float alpha = 1.0f - (t[j] - EPS);
        float w = alpha * run;                      // weights[i]
        run *= t[j];
        Wmat[ray * N_SAMP + i] = w;
        ir  += w * rgb[3 * i + 0];
        ig  += w * rgb[3 * i + 1];
        ib  += w * rgb[3 * i + 2];
        idp += w * invz[i];
    }
    // wave reduction (sum over all 32 lanes)
#pragma unroll
    for (int m = 16; m >= 1; m >>= 1) {
        ir  += __shfl_xor(ir,  m, 32);
        ig  += __shfl_xor(ig,  m, 32);
        ib  += __shfl_xor(ib,  m, 32);
        idp += __shfl_xor(idp, m, 32);
    }
    if (lane == 0) {
        out[ray * 3 + 0]     = ir;
        out[ray * 3 + 1]     = ig;
        out[ray * 3 + 2]     = ib;
        out[3 * N_RAYS + ray] = idp;                // invdepth block
    }
}

// ---------------------------------------------------------------------------
// Kernel 3: distortion via WMMA.  Each wave computes one 16x16 tile of
// P = W @ S for one 16-ray group (S tile built from z_lin in LDS), then the
// row-dot with W and writes per-(ray, colTile) partials (no atomics ->
// deterministic).  blockDim = 128 (4 waves), gridDim = 576 -> 2304 waves.
//
// Fragment layouts per CDNA5 ISA 7.12.2 (32-bit, wave32):
//   A 16x4 : lane L, M = L%16, VGPR v holds K = 2*(L/16) + v
//   B 4x16 : lane L, N = L%16, VGPR v holds K = 2*(L/16) + v
//   C 16x16: lane L, N = L%16, VGPR v holds M = v + 8*(L/16)
// ---------------------------------------------------------------------------
__global__ void k_dist(const float* __restrict__ Wmat,
                       const float* __restrict__ ws,
                       float* __restrict__ part) {
    __shared__ float zl[N_SAMP];
    const float* zlin = ws + WS_ZLIN;
    for (int i = threadIdx.x; i < N_SAMP; i += 128) zl[i] = zlin[i];
    __syncthreads();

    const int lane = threadIdx.x & 31;
    const int idx  = blockIdx.x * 4 + (threadIdx.x >> 5);   // 0..2303
    const int g    = idx / N_TILES;                         // ray group
    const int tile = idx % N_TILES;                         // 16-col tile of S

    const int m16  = lane & 15;
    const int h    = lane >> 4;                             // lane half
    const int col  = tile * 16 + m16;                       // B/C column
    const float zc = zl[col];
    const float* Arow = Wmat + (g * 16 + m16) * N_SAMP;     // A row for this lane

    v8f c = {};
    for (int k = 0; k < N_SAMP; k += 4) {
        const int kk = k + 2 * h;
        v2f a, b;
        a.x = Arow[kk];                                     // global_load_b64
        a.y = Arow[kk + 1];
        b.x = fabsf(zl[kk]     - zc);                       // S tile on the fly
        b.y = fabsf(zl[kk + 1] - zc);
        c = __builtin_amdgcn_wmma_f32_16x16x4_f32(
                /*neg_a=*/false, a, /*neg_b=*/false, b,
                /*c_mod=*/(short)0, c, /*reuse_a=*/false, /*reuse_b=*/false);
    }

    // l-partials: for each accumulator row, dot P[row, tileCols] with W.
#pragma unroll
    for (int v = 0; v < 8; ++v) {
        const int row = v + 8 * h;                          // M within group
        const int ray = g * 16 + row;
        float s = c[v] * Wmat[ray * N_SAMP + col];
        // reduce over the 16 lanes of this half (distinct N, same row set)
#pragma unroll
        for (int m = 8; m >= 1; m >>= 1) s += __shfl_xor(s, m, 16);
        if (m16 == 0) part[ray * N_TILES + tile] = s;
    }
}

// ---------------------------------------------------------------------------
// Kernel 4: l_dist = mean over rays of sum over tiles.
// ---------------------------------------------------------------------------
__global__ void k_final(const float* __restrict__ part, float* __restrict__ out) {
    __shared__ float sm[256];
    float s = 0.0f;
    for (int r = threadIdx.x; r < N_RAYS; r += 256) {
        const float* p = part + r * N_TILES;
        float l = 0.0f;
#pragma unroll
        for (int t = 0; t < N_TILES; ++t) l += p[t];
        s += l;
    }
    sm[threadIdx.x] = s;
    __syncthreads();
    for (int o = 128; o > 0; o >>= 1) {
        if (threadIdx.x < o) sm[threadIdx.x] += sm[threadIdx.x + o];
        __syncthreads();
    }
    if (threadIdx.x == 0) out[4 * N_RAYS] = sm[0] * (1.0f / (float)N_RAYS);
}

// ---------------------------------------------------------------------------
extern "C" void kernel_launch(void* const* d_in, const int* in_sizes, int n_in,
                              void* d_out, int out_size, void* d_ws, size_t ws_size,
                              hipStream_t stream) {
    const float* sigmas = (const float*)d_in[0];
    const float* rgbs   = (const float*)d_in[1];
    float* out = (float*)d_out;
    float* ws  = (float*)d_ws;

    float* Wmat = ws + WS_W;
    float* part = ws + WS_PART;

    k_init   <<<1,   256, 0, stream>>>(ws);
    k_weights<<<128, 256, 0, stream>>>(sigmas, rgbs, ws, Wmat, out);
    k_dist   <<<576, 128, 0, stream>>>(Wmat, ws, part);
    k_final  <<<1,   256, 0, stream>>>(part, out);
}